// MSDeformableAttention_17841294147884
// MI455X (gfx1250) — compile-verified
//
#include <hip/hip_runtime.h>
#include <hip/hip_bf16.h>

typedef __attribute__((ext_vector_type(2))) float v2f;
typedef __attribute__((ext_vector_type(8))) float v8f;

#define BS 16
#define LQ 300
#define EMBED 256
#define NHEADS 8
#define HEAD_DIM 32
#define SUM_POINTS 16
#define LV 8500

// ---------------------------------------------------------------------------
// Kernel 1: fp32 GEMM  C[M,N] = A[M,K] @ B[K,N] + bias[N]
// One wave computes one 16x16 tile via V_WMMA_F32_16X16X4_F32.
// A-frag (16x4): lanes 0-15 -> M=lane, VGPR0/1 = K=k,k+1 (float2 load);
//                lanes 16-31 -> M=lane-16, K=k+2,k+3.
// B-frag (4x16): lanes 0-15 -> N=lane, K=k,k+1; lanes 16-31 -> K=k+2,k+3.
// C (16x16 f32): VGPR v, lanes 0-15 -> row v, col lane; lanes 16-31 -> row v+8.
// ---------------------------------------------------------------------------
__global__ __launch_bounds__(128) void gemm_wmma_f32(
    const float* __restrict__ A, const float* __restrict__ B,
    const float* __restrict__ bias, float* __restrict__ C,
    int M, int N, int K)
{
  int wave = (int)((blockIdx.x * blockDim.x + threadIdx.x) >> 5);
  int lane = (int)(threadIdx.x & 31);
  int ntiles = N >> 4;
  int mt = wave / ntiles;
  int nt = wave - mt * ntiles;
  if (mt * 16 >= M) return;

  int lane15 = lane & 15;
  int khalf  = (lane >> 4) * 2;                 // 0 for lanes 0-15, 2 for 16-31
  int arow   = mt * 16 + lane15;
  int col    = nt * 16 + lane15;

  const float* Arow = A + (size_t)arow * K + khalf;
  const float* Bcol = B + (size_t)khalf * N + col;

  v8f acc = {};
#pragma unroll 4
  for (int k = 0; k < K; k += 4) {
    v2f a = *(const v2f*)(Arow + k);            // A[arow][k+khalf .. k+khalf+1]
    v2f b;
    b.x = Bcol[(size_t)k * N];                  // B[k+khalf]  [col]
    b.y = Bcol[(size_t)(k + 1) * N];            // B[k+khalf+1][col]
    acc = __builtin_amdgcn_wmma_f32_16x16x4_f32(
        /*neg_a=*/false, a, /*neg_b=*/false, b,
        /*c_mod=*/(short)0, acc, /*reuse_a=*/false, /*reuse_b=*/false);
  }

  float bv = bias[col];
  int rbase = mt * 16 + (lane >> 4) * 8;        // rows rbase..rbase+7
  float* Cp = C + (size_t)rbase * N + col;
#pragma unroll
  for (int v = 0; v < 8; ++v)
    Cp[(size_t)v * N] = acc[v] + bv;
}

// ---------------------------------------------------------------------------
// Kernel 2: softmax over the 16 points of each (b, q, h) row, in place.
// ---------------------------------------------------------------------------
__global__ __launch_bounds__(256) void softmax16(float* __restrict__ attn, int rows)
{
  int t = (int)(blockIdx.x * blockDim.x + threadIdx.x);
  if (t >= rows) return;
  float* p = attn + (size_t)t * SUM_POINTS;
  float v[SUM_POINTS];
  float m = -__builtin_inff();
#pragma unroll
  for (int i = 0; i < SUM_POINTS; ++i) { v[i] = p[i]; m = fmaxf(m, v[i]); }
  float s = 0.f;
#pragma unroll
  for (int i = 0; i < SUM_POINTS; ++i) { v[i] = __expf(v[i] - m); s += v[i]; }
  float r = 1.0f / s;
#pragma unroll
  for (int i = 0; i < SUM_POINTS; ++i) p[i] = v[i] * r;
}

// ---------------------------------------------------------------------------
// Kernel 3: bilinear sampling + weighted accumulation.
// Block = (b,q); wave = head; lane = channel (HEAD_DIM == 32 == wave32).
// Location math is wave-uniform; each lane gathers its own channel.
//   pixel x = loc_x * W - 0.5  (simplified grid_sample, align_corners=False)
// ---------------------------------------------------------------------------
__global__ __launch_bounds__(256) void ms_deform_sample(
    const float* __restrict__ value, const float* __restrict__ refp,
    const float* __restrict__ off,   const float* __restrict__ attn,
    const float* __restrict__ nps,   float* __restrict__ out)
{
  const int lH[4] = {80, 40, 20, 10};
  const int lW[4] = {80, 40, 20, 10};
  const int lO[4] = {0, 6400, 8000, 8400};

  int bq = (int)blockIdx.x;                     // 0..BS*LQ-1
  int h  = (int)(threadIdx.x >> 5);             // 0..7
  int c  = (int)(threadIdx.x & 31);             // channel
  int b  = bq / LQ;

  const float* offr = off  + (size_t)bq * (NHEADS * SUM_POINTS * 2) + h * (SUM_POINTS * 2);
  const float* attr = attn + (size_t)bq * (NHEADS * SUM_POINTS)     + h * SUM_POINTS;
  const float* r    = refp + (size_t)bq * 4;
  float rcx = r[0], rcy = r[1], rw = r[2], rh = r[3];

  const float* v = value + ((size_t)(b * NHEADS + h) * HEAD_DIM + c) * LV;

  float acc = 0.f;
#pragma unroll
  for (int p = 0; p < SUM_POINTS; ++p) {
    int lvl  = p >> 2;
    int Hh   = lH[lvl], Ww = lW[lvl], base = lO[lvl];
    float s  = nps[p] * 0.5f;                   // num_points_scale * OFFSET_SCALE
    float lx = rcx + offr[2 * p]     * s * rw;
    float ly = rcy + offr[2 * p + 1] * s * rh;
    float x  = lx * (float)Ww - 0.5f;
    float y  = ly * (float)Hh - 0.5f;
    float x0f = floorf(x), y0f = floorf(y);
    float wx1 = x - x0f,   wy1 = y - y0f;
    float wx0 = 1.f - wx1, wy0 = 1.f - wy1;
    int x0 = (int)x0f, y0 = (int)y0f;
    float a = attr[p];

    auto corner = [&](int xi, int yi, float w) -> float {
      bool valid = (xi >= 0) & (xi <= Ww - 1) & (yi >= 0) & (yi <= Hh - 1);
      int xc = min(max(xi, 0), Ww - 1);
      int yc = min(max(yi, 0), Hh - 1);
      float g = v[base + yc * Ww + xc];
      return valid ? g * w : 0.f;
    };

    float sval = corner(x0,     y0,     wx0 * wy0)
               + corner(x0 + 1, y0,     wx1 * wy0)
               + corner(x0,     y0 + 1, wx0 * wy1)
               + corner(x0 + 1, y0 + 1, wx1 * wy1);
    acc += a * sval;
  }

  out[(size_t)bq * EMBED + h * HEAD_DIM + c] = acc;
}

// ---------------------------------------------------------------------------
extern "C" void kernel_launch(void* const* d_in, const int* in_sizes, int n_in,
                              void* d_out, int out_size, void* d_ws, size_t ws_size,
                              hipStream_t stream) {
  const float* query  = (const float*)d_in[0];
  const float* refp   = (const float*)d_in[1];
  const float* value  = (const float*)d_in[2];
  // d_in[3] = value_spatial_shapes (compile-time constants, unused)
  const float* W_off  = (const float*)d_in[4];
  const float* b_off  = (const float*)d_in[5];
  const float* W_attn = (const float*)d_in[6];
  const float* b_attn = (const float*)d_in[7];
  const float* nps    = (const float*)d_in[8];
  float* out = (float*)d_out;

  const int M = BS * LQ;                         // 4800
  float* off_ws  = (float*)d_ws;                 // M x 256
  float* attn_ws = off_ws + (size_t)M * 256;     // M x 128

  {   // offsets GEMM: N = 256 -> (300 * 16) waves
    int waves  = (M / 16) * (256 / 16);
    int blocks = (waves * 32) / 128;
    gemm_wmma_f32<<<blocks, 128, 0, stream>>>(query, W_off, b_off, off_ws, M, 256, EMBED);
  }
  {   // attention-logits GEMM: N = 128 -> (300 * 8) waves
    int waves  = (M / 16) * (128 / 16);
    int blocks = (waves * 32) / 128;
    gemm_wmma_f32<<<blocks, 128, 0, stream>>>(query, W_attn, b_attn, attn_ws, M, 128, EMBED);
  }
  {   // softmax over 16 points for each of M*NHEADS rows
    int rows = M * NHEADS;                       // 38400
    softmax16<<<(rows + 255) / 256, 256, 0, stream>>>(attn_ws, rows);
  }
  {   // sampling: block per (b,q), wave per head, lane per channel
    ms_deform_sample<<<M, 256, 0, stream>>>(value, refp, off_ws, attn_ws, nps, out);
  }
}